// GLMNB_17008070492498
// MI455X (gfx1250) — compile-verified
//
#include <hip/hip_runtime.h>
#include <math.h>

typedef __attribute__((ext_vector_type(2))) float v2f;
typedef __attribute__((ext_vector_type(8))) float v8f;

#define NVOX  300000
#define BDIM  512
#define NSTUD 2000

// ---------------------------------------------------------------------------
// Setup: moderator reductions + all scalar constants (incl. lgamma terms,
// since r = v*sum_mu^2/sum_mu2 is voxel-independent).
// cst layout: [0]=sum_mu [1]=sum_mu2 [2]=r [3..8]=tt[y] y=0..5
//             [9]=v [10]=log(v*sum_mu) [11]=v*sum_mu
// ---------------------------------------------------------------------------
__global__ __launch_bounds__(256) void glm_setup(
    const float* __restrict__ mod, const float* __restrict__ gamma,
    const float* __restrict__ alpha_sqrt, float* __restrict__ cst)
{
    __shared__ float w1[8], w2[8];
    const float g0 = gamma[0], g1 = gamma[1], g2 = gamma[2];
    float s1 = 0.f, s2 = 0.f;
    for (int s = (int)threadIdx.x; s < NSTUD; s += 256) {
        float d = fmaf(mod[3 * s + 2], g2, fmaf(mod[3 * s + 1], g1, mod[3 * s] * g0));
        float m = expf(d);
        s1 += m; s2 += m * m;
    }
    for (int off = 16; off > 0; off >>= 1) {
        s1 += __shfl_down(s1, off, 32);
        s2 += __shfl_down(s2, off, 32);
    }
    const int lane = threadIdx.x & 31, wave = threadIdx.x >> 5;
    if (lane == 0) { w1[wave] = s1; w2[wave] = s2; }
    __syncthreads();
    if (threadIdx.x == 0) {
        float A = 0.f, B2 = 0.f;
        for (int i = 0; i < 8; ++i) { A += w1[i]; B2 += w2[i]; }
        const float asq   = alpha_sqrt[0];
        const float v     = 1.f / (asq * asq);
        const float r     = v * A * A / B2;
        const float lgr   = lgammaf(r);
        cst[0] = A; cst[1] = B2; cst[2] = r;
        cst[3] = 0.f;
        for (int y = 1; y <= 5; ++y)
            cst[3 + y] = lgammaf((float)y + r) - lgammaf((float)y + 1.f) - lgr;
        cst[9]  = v;
        cst[10] = logf(v * A);
        cst[11] = v * A;
    }
}

// ---------------------------------------------------------------------------
// Main streaming pass: WMMA matvec (16 rows / wave, K in steps of 4) + the
// elementwise NB log-likelihood tail + reduction into out[0] (= -log_l).
// ---------------------------------------------------------------------------
__global__ __launch_bounds__(256) void glm_main(
    const float* __restrict__ coef, const float* __restrict__ beta,
    const int* __restrict__ foci, const float* __restrict__ cst,
    float* __restrict__ out)
{
    __shared__ float s_beta[BDIM];
    for (int i = (int)threadIdx.x; i < BDIM; i += 256) s_beta[i] = beta[i];
    __syncthreads();

    const int lane  = threadIdx.x & 31;
    const int wave  = threadIdx.x >> 5;
    const int group = blockIdx.x * 8 + wave;   // 16-row group index
    if (group * 16 >= NVOX) return;
    const int rowBase = group * 16;

    // A 16x4 f32 layout: lane L -> row M=L%16; VGPR0 K=(L/16)*2, VGPR1 K=+1.
    // B 4x16 f32 layout: same K striping; broadcast beta along N so every
    // D column carries the per-row dot product.
    const int m   = lane & 15;
    const int kof = (lane >> 4) << 1;          // 0 or 2
    const float* __restrict__ arow = coef + (size_t)(rowBase + m) * BDIM + kof;
    const float* __restrict__ brow = s_beta + kof;

    v8f acc = {};
#pragma unroll 8
    for (int k = 0; k < BDIM; k += 4) {
        v2f a = *(const v2f*)(arow + k);       // global_load_b64, 8B aligned
        v2f b = *(const v2f*)(brow + k);       // ds_load_b64 broadcast
        acc = __builtin_amdgcn_wmma_f32_16x16x4_f32(
            /*neg_a=*/false, a, /*neg_b=*/false, b,
            /*c_mod=*/(short)0, acc, /*reuse_a=*/false, /*reuse_b=*/false);
    }

    // D layout: lane L holds rows M=(L/16)*8 + j in acc[j].  Select acc[lane&7]
    // with a mux tree (avoids dynamic vector indexing / scratch).
    const int j = lane & 7;
    float a0 = (j & 4) ? acc[4] : acc[0];
    float a1 = (j & 4) ? acc[5] : acc[1];
    float a2 = (j & 4) ? acc[6] : acc[2];
    float a3 = (j & 4) ? acc[7] : acc[3];
    float b0 = (j & 2) ? a2 : a0;
    float b1 = (j & 2) ? a3 : a1;
    float dot = (j & 1) ? b1 : b0;

    // Lanes {0..7, 16..23} each own one unique voxel of the 16-row group.
    float term = 0.f;
    if ((lane & 15) < 8) {
        const float B2  = cst[1];
        const float r   = cst[2];
        const float lvA = cst[10];
        const float vA  = cst[11];
        const int voxel = rowBase + ((lane >> 4) << 3) + j;
        const int y     = foci[voxel];                 // y in [0,5]
        const float mu  = expf(dot);
        const float t   = mu * B2;                     // mu * sum_mu2
        const float logden = logf(vA + t);
        // r*log1p(-p) + y*log(p) + [y>0] (lgamma(y+r)-lgamma(y+1)-lgamma(r))
        term = r * (lvA - logden) + (float)y * (logf(t) - logden) + cst[3 + y];
    }
    for (int off = 16; off > 0; off >>= 1) term += __shfl_down(term, off, 32);
    if (lane == 0) atomicAdd(out, -term);
}

// ---------------------------------------------------------------------------
extern "C" void kernel_launch(void* const* d_in, const int* in_sizes, int n_in,
                              void* d_out, int out_size, void* d_ws, size_t ws_size,
                              hipStream_t stream) {
    (void)in_sizes; (void)n_in; (void)out_size; (void)ws_size;
    const float* coef  = (const float*)d_in[0];   // [V,B]
    const float* mod   = (const float*)d_in[1];   // [S,3]
    const int*   foci  = (const int*)  d_in[2];   // [V,1]
    const float* beta  = (const float*)d_in[3];   // [1,B]
    const float* gamma = (const float*)d_in[4];   // [1,3]
    const float* asq   = (const float*)d_in[5];   // scalar
    float* out = (float*)d_out;
    float* cst = (float*)d_ws;

    hipMemsetAsync(out, 0, sizeof(float), stream);
    glm_setup<<<1, 256, 0, stream>>>(mod, gamma, asq, cst);

    const int groups = (NVOX + 15) / 16;          // 18750
    const int blocks = (groups + 7) / 8;          // 2344
    glm_main<<<blocks, 256, 0, stream>>>(coef, beta, foci, cst, out);
}